// PredictAverageReward_16587163697300
// MI455X (gfx1250) — compile-verified
//
#include <hip/hip_runtime.h>
#include <hip/hip_bf16.h>

typedef __attribute__((ext_vector_type(2))) float v2f;
typedef __attribute__((ext_vector_type(8))) float v8f;

#define WAVES_PER_BLOCK 4

// Generic f32 WMMA GEMM: C[M,N] = A[M,K] @ B  (+ addc)
//   BT=false : B is [K,N] row-major
//   BT=true  : B is [N,K] row-major (i.e. C = A @ B^T)
//   GATHER   : A row m is A[gatherA[m], :]
// One wave computes one 16x16 tile of C via V_WMMA_F32_16X16X4_F32,
// stepping K by 8 with two accumulators to hide the WMMA RAW hazard.
template <bool BT, bool GATHER>
__global__ void wmma_gemm_f32_kernel(const float* __restrict__ A,
                                     const float* __restrict__ B,
                                     float* __restrict__ C,
                                     int Mdim, int Ndim, int Kdim,
                                     const int* __restrict__ gatherA,
                                     const float* __restrict__ addp) {
    const int lane = threadIdx.x & 31;
    const int wave = threadIdx.x >> 5;
    const int tn   = (blockIdx.x * WAVES_PER_BLOCK + wave) * 16;  // N tile base
    const int tm   = blockIdx.y * 16;                             // M tile base
    if (tn >= Ndim || tm >= Mdim) return;  // wave-uniform exit: EXEC stays all-1s

    const int l15   = lane & 15;
    const int khalf = (lane >> 4) << 1;   // lanes 16-31 carry K+2 / K+3 (ISA 16x4 A layout)

    int arow = tm + l15;
    if (GATHER) arow = gatherA[arow];
    const float* __restrict__ Arow = A + (size_t)arow * (size_t)Kdim;

    const int ncol = tn + l15;
    const float* __restrict__ Brow = BT ? (B + (size_t)ncol * (size_t)Kdim) : nullptr;

    v8f acc0 = {};
    v8f acc1 = {};
    for (int k = 0; k < Kdim; k += 8) {
        v2f a0, a1, b0, b1;
        a0.x = Arow[k + khalf];
        a0.y = Arow[k + khalf + 1];
        a1.x = Arow[k + 4 + khalf];
        a1.y = Arow[k + 4 + khalf + 1];
        if (BT) {
            b0.x = Brow[k + khalf];
            b0.y = Brow[k + khalf + 1];
            b1.x = Brow[k + 4 + khalf];
            b1.y = Brow[k + 4 + khalf + 1];
        } else {
            b0.x = B[(size_t)(k + khalf) * Ndim + ncol];
            b0.y = B[(size_t)(k + khalf + 1) * Ndim + ncol];
            b1.x = B[(size_t)(k + 4 + khalf) * Ndim + ncol];
            b1.y = B[(size_t)(k + 4 + khalf + 1) * Ndim + ncol];
        }
        // 8 args: (neg_a, A, neg_b, B, c_mod, C, reuse_a, reuse_b)
        acc0 = __builtin_amdgcn_wmma_f32_16x16x4_f32(false, a0, false, b0,
                                                     (short)0, acc0, false, false);
        acc1 = __builtin_amdgcn_wmma_f32_16x16x4_f32(false, a1, false, b1,
                                                     (short)0, acc1, false, false);
    }

    const float addc  = addp ? addp[0] : 0.0f;
    const int   mbase = tm + ((lane >> 4) << 3);  // C layout: VGPR r -> M=r / M=r+8
#pragma unroll
    for (int r = 0; r < 8; ++r)
        C[(size_t)(mbase + r) * Ndim + ncol] = acc0[r] + acc1[r] + addc;
}

// Dm[i,j] = (1/4096) * sum_f (Rd[f,i] >= Rd[f,j]);  Rd is [4096,256].
// Block = 256 threads = one 16x16 (i,j) tile; LDS-tiled over fruits.
__global__ void dm_kernel(const float* __restrict__ Rd, float* __restrict__ Dm) {
    __shared__ float si[16][17];
    __shared__ float sj[16][17];
    const int tid = threadIdx.x;
    const int li  = tid & 15;   // i within tile
    const int lf  = tid >> 4;   // j within tile (also f-row during loads)
    const int i0  = blockIdx.x * 16;
    const int j0  = blockIdx.y * 16;
    int cnt = 0;
    for (int f0 = 0; f0 < 4096; f0 += 16) {
        si[lf][li] = Rd[(size_t)(f0 + lf) * 256 + i0 + li];
        sj[lf][li] = Rd[(size_t)(f0 + lf) * 256 + j0 + li];
        __syncthreads();
#pragma unroll
        for (int f = 0; f < 16; ++f)
            cnt += (si[f][li] >= sj[f][lf]) ? 1 : 0;
        __syncthreads();
    }
    Dm[(size_t)(i0 + li) * 256 + (j0 + lf)] = (float)cnt * (1.0f / 4096.0f);
}

// Reference semantics: comp_K = triu(Dm,1) + triu(1-Dm,1).T ; diagonal = 0.
// choice = (comp >= 0.5) ? 0 : 1 ; proposal = [1, onehot3(choice)]  -> [4096,4]
__global__ void predict_kernel(const int* __restrict__ labels,
                               const float* __restrict__ Dm,
                               float* __restrict__ out) {
    const int b = blockIdx.x * blockDim.x + threadIdx.x;
    if (b >= 4096) return;
    const int t1 = labels[b];
    const int t2 = labels[4096 + b];
    float comp;
    if (t1 < t2)      comp = Dm[t1 * 256 + t2];
    else if (t1 > t2) comp = 1.0f - Dm[t2 * 256 + t1];
    else              comp = 0.0f;
    const int choice = (comp >= 0.5f) ? 0 : 1;
    float4 o;
    o.x = 1.0f;
    o.y = (choice == 0) ? 1.0f : 0.0f;
    o.z = (choice == 1) ? 1.0f : 0.0f;
    o.w = 0.0f;
    reinterpret_cast<float4*>(out)[b] = o;
}

extern "C" void kernel_launch(void* const* d_in, const int* in_sizes, int n_in,
                              void* d_out, int out_size, void* d_ws, size_t ws_size,
                              hipStream_t stream) {
    const float* fruits = (const float*)d_in[0];  // [8192,128]
    const float* tools  = (const float*)d_in[1];  // [1024,128]
    const float* Mf     = (const float*)d_in[2];  // [128,512]
    const float* Mt     = (const float*)d_in[3];  // [128,512]
    const float* Mm     = (const float*)d_in[4];  // [512,512]
    const float* minr   = (const float*)d_in[5];  // [1]
    const int*   domf   = (const int*)d_in[6];    // [4096]
    // d_in[7] = domain_t = arange(256): identity, folded into indexing.
    const int*   labels = (const int*)d_in[8];    // [2,4096]

    float* ws = (float*)d_ws;
    float* Pt = ws;               // [256,512]   tools_prop[0:256] @ M_tool
    float* T  = ws + 131072;      // [256,512]   Pt @ M
    float* Pf = ws + 262144;      // [4096,512]  fruits_prop[domain_f] @ M_fruit
    float* Rd = ws + 2359296;     // [4096,256]  Pf @ T^T + min_r
    float* Dm = ws + 3407872;     // [256,256]
    (void)ws_size; (void)in_sizes; (void)n_in; (void)out_size;

    const dim3 blk(32 * WAVES_PER_BLOCK);

    // 1) Pt = tools_prop[0:256] @ M_tool   (M=256, N=512, K=128)
    wmma_gemm_f32_kernel<false, false><<<dim3(512 / 64, 256 / 16), blk, 0, stream>>>(
        tools, Mt, Pt, 256, 512, 128, nullptr, nullptr);

    // 2) T = Pt @ M                        (M=256, N=512, K=512)
    wmma_gemm_f32_kernel<false, false><<<dim3(512 / 64, 256 / 16), blk, 0, stream>>>(
        Pt, Mm, T, 256, 512, 512, nullptr, nullptr);

    // 3) Pf = fruits_prop[domain_f] @ M_fruit   (M=4096, N=512, K=128, gathered A)
    wmma_gemm_f32_kernel<false, true><<<dim3(512 / 64, 4096 / 16), blk, 0, stream>>>(
        fruits, Mf, Pf, 4096, 512, 128, domf, nullptr);

    // 4) Rd = Pf @ T^T + min_r             (M=4096, N=256, K=512, B transposed)
    wmma_gemm_f32_kernel<true, false><<<dim3(256 / 64, 4096 / 16), blk, 0, stream>>>(
        Pf, T, Rd, 4096, 256, 512, nullptr, minr);

    // 5) Pairwise win-rate matrix
    dm_kernel<<<dim3(16, 16), dim3(256), 0, stream>>>(Rd, Dm);

    // 6) Batch prediction -> [4096,4]
    predict_kernel<<<dim3(16), dim3(256), 0, stream>>>(labels, Dm, (float*)d_out);
}